// BiViewMixHop_28492813041846
// MI455X (gfx1250) — compile-verified
//
#include <hip/hip_runtime.h>
#include <hip/hip_bf16.h>
#include <math.h>

// Problem constants (match reference)
#define NF 128   // node feature width (2H)
#define NH 64    // hidden per view
#define NG 256   // graphs
#define NC 10    // classes

typedef float v2f __attribute__((ext_vector_type(2)));
typedef float v8f __attribute__((ext_vector_type(8)));

// ---------------- utility kernels ----------------

__global__ void zero_kernel(float* __restrict__ p, int n) {
  int i = blockIdx.x * blockDim.x + threadIdx.x;
  if (i < n) p[i] = 0.0f;
}

__global__ void deg_kernel(const int* __restrict__ dst, const float* __restrict__ mask,
                           float* __restrict__ deg, int E) {
  int e = blockIdx.x * blockDim.x + threadIdx.x;
  if (e < E) atomicAdd(&deg[dst[e]], mask[e]);
}

__global__ void counts_kernel(const int* __restrict__ batch, float* __restrict__ counts, int n) {
  int i = blockIdx.x * blockDim.x + threadIdx.x;
  if (i < n) atomicAdd(&counts[batch[i]], 1.0f);
}

// One hop: hout += mask[e] * hin[src[e]]  (scatter into dst rows). 32 threads/edge, float4 each.
__global__ void scatter_kernel(const float* __restrict__ hin, const int* __restrict__ src,
                               const int* __restrict__ dst, const float* __restrict__ mask,
                               float* __restrict__ hout, int E) {
  int idx = blockIdx.x * blockDim.x + threadIdx.x;
  if (idx >= E * 32) return;
  int e = idx >> 5;
  int f = (idx & 31) << 2;
  float m = mask[e];
  int s = src[e], d = dst[e];
  float4 v = *(const float4*)(hin + (size_t)s * NF + f);
  float* o = hout + (size_t)d * NF + f;
  atomicAdd(o + 0, m * v.x);
  atomicAdd(o + 1, m * v.y);
  atomicAdd(o + 2, m * v.z);
  atomicAdd(o + 3, m * v.w);
}

__global__ void divdeg_kernel(float* __restrict__ h, const float* __restrict__ deg, int total) {
  int i = blockIdx.x * blockDim.x + threadIdx.x;
  if (i < total) h[i] /= fmaxf(deg[i >> 7], 1.0f);   // NF == 128
}

// ---------------- fused MixHop matmul: out[:,col..col+64) = relu(h0@W0 + h1@W1 + h2@W2 + b) ----
// One wave per 16x64 output tile, f32 WMMA 16x16x4, K = 128 (32 steps), W_j staged in LDS.

__global__ __launch_bounds__(128) void conv_wmma_kernel(
    const float* __restrict__ h0, const float* __restrict__ h1, const float* __restrict__ h2,
    const float* __restrict__ W,     // [3, 128, 64] row-major
    const float* __restrict__ bias,  // [64]
    float* __restrict__ out,         // [N, 128]
    int col_off, int nrowblocks) {
  __shared__ float sW[NF * NH];      // 32 KB: one hop matrix at a time

  const int lane = threadIdx.x & 31;
  const int wave = threadIdx.x >> 5;
  const int rb   = blockIdx.x * 4 + wave;       // 16-node row block
  const bool active = (rb < nrowblocks);

  const int m    = lane & 15;                   // M (A) / N (B,C,D) index
  const int koff = (lane >> 4) << 1;            // 0 or 2: K sub-offset per half-wave
  const int node_a = rb * 16 + m;

  v8f acc[4];
#pragma unroll
  for (int t = 0; t < 4; ++t) {
    float bv = bias[t * 16 + m];
#pragma unroll
    for (int r = 0; r < 8; ++r) acc[t][r] = bv;
  }

  for (int j = 0; j < 3; ++j) {
    __syncthreads();
    // stage W_j into LDS (2048 float4 loads across 128 threads)
    const float4* Wj = (const float4*)(W + (size_t)j * NF * NH);
    for (int i = threadIdx.x; i < (NF * NH) / 4; i += 128) ((float4*)sW)[i] = Wj[i];
    __syncthreads();

    if (active) {
      const float* hj = (j == 0) ? h0 : (j == 1) ? h1 : h2;
      const float* arow = hj + (size_t)node_a * NF + koff;
#pragma unroll 4
      for (int k0 = 0; k0 < NF; k0 += 4) {
        v2f a = *(const v2f*)(arow + k0);       // A: M=lane&15, K = k0+koff+{0,1}
#pragma unroll
        for (int t = 0; t < 4; ++t) {
          v2f b;                                 // B mirrors A striping: N=lane&15
          b.x = sW[(k0 + koff) * NH + t * 16 + m];
          b.y = sW[(k0 + koff + 1) * NH + t * 16 + m];
          acc[t] = __builtin_amdgcn_wmma_f32_16x16x4_f32(
              false, a, false, b, (short)0, acc[t], false, false);
        }
      }
    }
  }

  if (active) {
    const int half = lane >> 4;                  // C/D: lanes16-31 hold M = r+8
#pragma unroll
    for (int t = 0; t < 4; ++t) {
#pragma unroll
      for (int r = 0; r < 8; ++r) {
        int node = rb * 16 + r + half * 8;
        int col  = col_off + t * 16 + m;
        out[(size_t)node * NF + col] = fmaxf(acc[t][r], 0.0f);
      }
    }
  }
}

// ---------------- graph readout ----------------
// h is post-ReLU (>= 0), so uint-bit atomicMax == float max and 0-init matches the
// reference's "empty graph -> 0" fixup.

__global__ void pool_scatter_kernel(const float* __restrict__ h, const int* __restrict__ batch,
                                    float* __restrict__ maxbuf, float* __restrict__ sumbuf,
                                    int total) {
  int i = blockIdx.x * blockDim.x + threadIdx.x;
  if (i >= total) return;
  int n = i >> 7, f = i & 127;
  int g = batch[n];
  float v = h[i];
  atomicMax((unsigned int*)&maxbuf[g * NF + f], __float_as_uint(v));
  atomicAdd(&sumbuf[g * NF + f], v);
}

__global__ void pool_accum_kernel(float* __restrict__ r, const float* __restrict__ maxbuf,
                                  const float* __restrict__ sumbuf,
                                  const float* __restrict__ counts, int total) {
  int i = blockIdx.x * blockDim.x + threadIdx.x;
  if (i >= total) return;
  int g = i >> 7, f = i & 127;
  r[g * 256 + f]       += maxbuf[i];
  r[g * 256 + 128 + f] += sumbuf[i] / fmaxf(counts[g], 1.0f);
}

// ---------------- MLP head: 256 graphs, one block each ----------------

__global__ __launch_bounds__(128) void mlp_head_kernel(
    const float* __restrict__ r,
    const float* __restrict__ W1, const float* __restrict__ b1,   // [256,128],[128]
    const float* __restrict__ W2, const float* __restrict__ b2,   // [128,64],[64]
    const float* __restrict__ W3, const float* __restrict__ b3,   // [64,10],[10]
    float* __restrict__ out) {
  __shared__ float sr[256];
  __shared__ float s1[128];
  __shared__ float s2[64];
  __shared__ float sl[NC];
  __shared__ float red[2];
  int g = blockIdx.x;
  int t = threadIdx.x;

  sr[t]       = r[g * 256 + t];
  sr[t + 128] = r[g * 256 + 128 + t];
  __syncthreads();

  float acc = b1[t];
  for (int k = 0; k < 256; ++k) acc += sr[k] * W1[k * 128 + t];
  s1[t] = fmaxf(acc, 0.0f);
  __syncthreads();

  if (t < 64) {
    acc = b2[t];
    for (int k = 0; k < 128; ++k) acc += s1[k] * W2[k * 64 + t];
    s2[t] = fmaxf(acc, 0.0f);
  }
  __syncthreads();

  if (t < NC) {
    acc = b3[t];
    for (int k = 0; k < 64; ++k) acc += s2[k] * W3[k * NC + t];
    sl[t] = acc;
  }
  __syncthreads();

  if (t == 0) {
    float mx = sl[0];
    for (int c = 1; c < NC; ++c) mx = fmaxf(mx, sl[c]);
    float s = 0.0f;
    for (int c = 0; c < NC; ++c) s += expf(sl[c] - mx);
    red[0] = mx;
    red[1] = logf(s);
  }
  __syncthreads();
  if (t < NC) out[g * NC + t] = sl[t] - red[0] - red[1];
}

// ---------------- host orchestration ----------------

extern "C" void kernel_launch(void* const* d_in, const int* in_sizes, int n_in,
                              void* d_out, int out_size, void* d_ws, size_t ws_size,
                              hipStream_t stream) {
  const float* x        = (const float*)d_in[0];
  const int*   edge     = (const int*)d_in[1];
  const int*   batch    = (const int*)d_in[2];
  const float* hom_mask = (const float*)d_in[3];
  const float* het_mask = (const float*)d_in[4];
  const float* hom_W    = (const float*)d_in[5];   // [3,3,128,64]
  const float* hom_b    = (const float*)d_in[6];   // [3,64]
  const float* het_W    = (const float*)d_in[7];
  const float* het_b    = (const float*)d_in[8];
  const float* l1W = (const float*)d_in[9];
  const float* l1b = (const float*)d_in[10];
  const float* l2W = (const float*)d_in[11];
  const float* l2b = (const float*)d_in[12];
  const float* l3W = (const float*)d_in[13];
  const float* l3b = (const float*)d_in[14];
  float* out = (float*)d_out;

  const int N = in_sizes[0] / NF;      // 50000
  const int E = in_sizes[1] / 2;       // 800000
  const int NFh = N * NF;              // 6.4M floats
  const int nrb = N / 16;              // 3125 row blocks (N is a multiple of 16)

  const int* src = edge;
  const int* dst = edge + E;

  // workspace partition (floats)
  float* ws      = (float*)d_ws;
  float* hA      = ws;
  float* hB      = hA + (size_t)NFh;
  float* p1      = hB + (size_t)NFh;
  float* p2      = p1 + (size_t)NFh;
  float* deg_hom = p2 + (size_t)NFh;
  float* deg_het = deg_hom + N;
  float* counts  = deg_het + N;
  float* maxbuf  = counts + NG;
  float* sumbuf  = maxbuf + NG * NF;
  float* racc    = sumbuf + NG * NF;   // [G, 4H] = [256,256]

  const int TB = 256;
  auto nb = [](int n, int b) { return (n + b - 1) / b; };

  // degrees, graph counts, r = 0
  zero_kernel<<<nb(2 * N + NG, TB), TB, 0, stream>>>(deg_hom, 2 * N + NG);
  zero_kernel<<<nb(NG * 256, TB), TB, 0, stream>>>(racc, NG * 256);
  deg_kernel<<<nb(E, TB), TB, 0, stream>>>(dst, hom_mask, deg_hom, E);
  deg_kernel<<<nb(E, TB), TB, 0, stream>>>(dst, het_mask, deg_het, E);
  counts_kernel<<<nb(N, TB), TB, 0, stream>>>(batch, counts, N);

  auto conv = [&](const float* hin, const float* mask, const float* deg,
                  const float* W, const float* b, float* hout, int col) {
    zero_kernel<<<nb(NFh, TB), TB, 0, stream>>>(p1, NFh);
    scatter_kernel<<<nb(E * 32, TB), TB, 0, stream>>>(hin, src, dst, mask, p1, E);
    divdeg_kernel<<<nb(NFh, TB), TB, 0, stream>>>(p1, deg, NFh);
    zero_kernel<<<nb(NFh, TB), TB, 0, stream>>>(p2, NFh);
    scatter_kernel<<<nb(E * 32, TB), TB, 0, stream>>>(p1, src, dst, mask, p2, E);
    divdeg_kernel<<<nb(NFh, TB), TB, 0, stream>>>(p2, deg, NFh);
    conv_wmma_kernel<<<nb(nrb, 4), 128, 0, stream>>>(hin, p1, p2, W, b, hout, col, nrb);
  };

  auto readout = [&](const float* h) {
    zero_kernel<<<nb(2 * NG * NF, TB), TB, 0, stream>>>(maxbuf, 2 * NG * NF);
    pool_scatter_kernel<<<nb(NFh, TB), TB, 0, stream>>>(h, batch, maxbuf, sumbuf, NFh);
    pool_accum_kernel<<<nb(NG * NF, TB), TB, 0, stream>>>(racc, maxbuf, sumbuf, counts, NG * NF);
  };

  const size_t WL = 3 * (size_t)NF * NH;  // weights per layer per view

  // layer 0: x -> hA
  conv(x, hom_mask, deg_hom, hom_W + 0 * WL, hom_b + 0 * NH, hA, 0);
  conv(x, het_mask, deg_het, het_W + 0 * WL, het_b + 0 * NH, hA, NH);
  // layer 1: hA -> hB, readout(hB)
  conv(hA, hom_mask, deg_hom, hom_W + 1 * WL, hom_b + 1 * NH, hB, 0);
  conv(hA, het_mask, deg_het, het_W + 1 * WL, het_b + 1 * NH, hB, NH);
  readout(hB);
  // layer 2: hB -> hA, readout(hA)
  conv(hB, hom_mask, deg_hom, hom_W + 2 * WL, hom_b + 2 * NH, hA, 0);
  conv(hB, het_mask, deg_het, het_W + 2 * WL, het_b + 2 * NH, hA, NH);
  readout(hA);

  // MLP head + log_softmax
  mlp_head_kernel<<<NG, 128, 0, stream>>>(racc, l1W, l1b, l2W, l2b, l3W, l3b, out);
}